// TopKSAE_57896159150392
// MI455X (gfx1250) — compile-verified
//
#include <hip/hip_runtime.h>
#include <hip/hip_bf16.h>
#include <stdint.h>

typedef float v2f __attribute__((ext_vector_type(2)));
typedef float v8f __attribute__((ext_vector_type(8)));

#define N_ROWS  8192
#define DM      1024
#define DICT    16384
#define TOPK    256

// ---------------------------------------------------------------------------
// CDNA5 async global->LDS copy (ASYNCcnt-tracked DMA, bypasses VGPRs).
// lds_off: byte offset within workgroup LDS (low 32 bits of generic pointer).
// ---------------------------------------------------------------------------
__device__ __forceinline__ void async_copy_b128(void* lds_dst, const void* gsrc) {
    const uint32_t lds_off = (uint32_t)(uintptr_t)lds_dst;
    asm volatile("global_load_async_to_lds_b128 %0, %1, off"
                 :: "v"(lds_off), "v"((uint64_t)(uintptr_t)gsrc)
                 : "memory");
}

// ---------------------------------------------------------------------------
// Transpose W_dec [DM][DICT] -> WdT [DICT][DM]  (64MB, stays L2-resident)
// ---------------------------------------------------------------------------
__global__ __launch_bounds__(256) void k_transpose(const float* __restrict__ in,
                                                   float* __restrict__ out) {
    __shared__ float tile[32][33];
    const int kBase = blockIdx.x * 32;
    const int dBase = blockIdx.y * 32;
    const int tx = threadIdx.x;   // 0..31
    const int ty = threadIdx.y;   // 0..7
#pragma unroll
    for (int j = 0; j < 4; ++j) {
        const int d = dBase + ty + j * 8;
        tile[ty + j * 8][tx] = in[(size_t)d * DICT + kBase + tx];
    }
    __syncthreads();
#pragma unroll
    for (int j = 0; j < 4; ++j) {
        const int k = kBase + ty + j * 8;
        out[(size_t)k * DM + dBase + tx] = tile[tx][ty + j * 8];
    }
}

// ---------------------------------------------------------------------------
// Encoder GEMM: pre_acts[m, n] = sum_d X[m,d] * W[n,d] + bias[n]
// Block tile 128(M) x 128(N), 8 waves, wave tile 64x32 (4x2 WMMA 16x16 tiles).
// K staged in LDS chunks of 16 via DOUBLE-BUFFERED async global->LDS DMA.
// ---------------------------------------------------------------------------
__global__ __launch_bounds__(256) void k_encoder(const float* __restrict__ X,
                                                 const float* __restrict__ W,
                                                 const float* __restrict__ bias,
                                                 float* __restrict__ out) {
    __shared__ float xs[2][128][18];   // pad 18 -> conflict-free strided reads
    __shared__ float wsh[2][128][18];

    const int tid   = threadIdx.x;
    const int lane  = tid & 31;
    const int wave  = tid >> 5;       // 0..7
    const int waveM = wave >> 2;      // 0..1  -> 64 rows each
    const int waveN = wave & 3;       // 0..3  -> 32 cols each
    const int h     = lane >> 4;      // lane half
    const int r     = lane & 15;

    const int mBase = blockIdx.y * 128;
    const int nBase = blockIdx.x * 128;

    // each thread owns 2 float4 slots per tile: t in {tid, tid+256} of 512
    const int row0 = tid >> 2;                 // 0..63
    const int row1 = (tid + 256) >> 2;         // 64..127
    const int c40  = (tid & 3) << 2;           // 0/4/8/12
    const float* xRow0 = &X[(size_t)(mBase + row0) * DM + c40];
    const float* xRow1 = &X[(size_t)(mBase + row1) * DM + c40];
    const float* wRow0 = &W[(size_t)(nBase + row0) * DM + c40];
    const float* wRow1 = &W[(size_t)(nBase + row1) * DM + c40];

    v8f acc[4][2];
#pragma unroll
    for (int mi = 0; mi < 4; ++mi)
#pragma unroll
        for (int ni = 0; ni < 2; ++ni)
            acc[mi][ni] = (v8f)(0.0f);

    // prologue: async-stage K-chunk 0 into buffer 0 (4 b128 per thread)
    async_copy_b128(&xs[0][row0][c40], xRow0);
    async_copy_b128(&xs[0][row1][c40], xRow1);
    async_copy_b128(&wsh[0][row0][c40], wRow0);
    async_copy_b128(&wsh[0][row1][c40], wRow1);

    int buf = 0;
    for (int kb = 0; kb < DM; kb += 16) {
        const bool more = (kb + 16) < DM;
        if (more) {
            const int nb = buf ^ 1;
            const int ko = kb + 16;
            async_copy_b128(&xs[nb][row0][c40], xRow0 + ko);
            async_copy_b128(&xs[nb][row1][c40], xRow1 + ko);
            async_copy_b128(&wsh[nb][row0][c40], wRow0 + ko);
            async_copy_b128(&wsh[nb][row1][c40], wRow1 + ko);
            // 8 outstanding; oldest 4 (current buffer) must land
            asm volatile("s_wait_asynccnt 4" ::: "memory");
        } else {
            asm volatile("s_wait_asynccnt 0" ::: "memory");
        }
        __syncthreads();   // current buffer visible to all waves

#pragma unroll
        for (int kk = 0; kk < 4; ++kk) {
            const int kc = kk * 4 + 2 * h;   // K = 2h + j  (ISA A/B layout)
            v2f a[4], b[2];
#pragma unroll
            for (int mi = 0; mi < 4; ++mi)
                a[mi] = *(const v2f*)&xs[buf][waveM * 64 + mi * 16 + r][kc];
#pragma unroll
            for (int ni = 0; ni < 2; ++ni)
                b[ni] = *(const v2f*)&wsh[buf][waveN * 32 + ni * 16 + r][kc];
#pragma unroll
            for (int mi = 0; mi < 4; ++mi)
#pragma unroll
                for (int ni = 0; ni < 2; ++ni)
                    acc[mi][ni] = __builtin_amdgcn_wmma_f32_16x16x4_f32(
                        false, a[mi], false, b[ni], (short)0, acc[mi][ni],
                        false, false);
        }
        __syncthreads();   // all waves done reading buf before it is refilled
        buf ^= 1;
    }

    // epilogue: add bias, store (C/D layout: M = j + 8h, N = r)
#pragma unroll
    for (int ni = 0; ni < 2; ++ni) {
        const int col = nBase + waveN * 32 + ni * 16 + r;
        const float bv = bias[col];
#pragma unroll
        for (int mi = 0; mi < 4; ++mi) {
#pragma unroll
            for (int j = 0; j < 8; ++j) {
                const int rowg = mBase + waveM * 64 + mi * 16 + j + 8 * h;
                out[(size_t)rowg * DICT + col] = acc[mi][ni][j] + bv;
            }
        }
    }
}

// ---------------------------------------------------------------------------
// Per-row exact top-256 radix select. Row (64KB) lives in LDS.
// ---------------------------------------------------------------------------
__device__ __forceinline__ uint32_t sortable_key(float v) {
    uint32_t u = __float_as_uint(v);
    const uint32_t mask = (u & 0x80000000u) ? 0xFFFFFFFFu : 0x80000000u;
    return u ^ mask;   // monotonic: larger float -> larger key
}

__global__ __launch_bounds__(512) void k_topk(float* __restrict__ z,
                                              uint32_t* __restrict__ idxList,
                                              float* __restrict__ valList,
                                              uint32_t* __restrict__ counts) {
    __shared__ float vals[DICT];          // 64KB (320KB/WGP on CDNA5)
    __shared__ uint32_t hist[256];
    __shared__ uint32_t sPrefix, sWant, sTie, sOut;

    const int tid = threadIdx.x;
    const int row = blockIdx.x;
    float* rowPtr = z + (size_t)row * DICT;

    for (int i = tid; i < DICT; i += 512) vals[i] = rowPtr[i];
    if (tid == 0) { sPrefix = 0u; sWant = TOPK; sTie = 0u; sOut = 0u; }
    __syncthreads();

    // 4 passes, byte 3 (MSB) down to byte 0: exact threshold key + tie budget
    for (int p = 3; p >= 0; --p) {
        if (tid < 256) hist[tid] = 0u;
        __syncthreads();
        const uint32_t prefix   = sPrefix;
        const uint32_t maskHigh = (p == 3) ? 0u : (0xFFFFFFFFu << (8 * (p + 1)));
        for (int i = tid; i < DICT; i += 512) {
            const uint32_t key = sortable_key(vals[i]);
            if (((key ^ prefix) & maskHigh) == 0u)
                atomicAdd(&hist[(key >> (8 * p)) & 255u], 1u);
        }
        __syncthreads();
        if (tid == 0) {
            uint32_t cum = 0, want = sWant;
            for (int b = 255; b >= 0; --b) {
                const uint32_t c = hist[b];
                if (cum + c >= want) {
                    sPrefix = prefix | ((uint32_t)b << (8 * p));
                    sWant   = want - cum;
                    break;
                }
                cum += c;
            }
        }
        __syncthreads();
    }

    const uint32_t T  = sPrefix;  // key of the 256th-largest element
    const uint32_t nt = sWant;    // number of ties (==T) to accept

    for (int i = tid; i < DICT; i += 512) {
        const float v = vals[i];
        const uint32_t key = sortable_key(v);
        bool sel = false;
        if (key > T) sel = true;
        else if (key == T) {
            const uint32_t s = atomicAdd(&sTie, 1u);
            sel = (s < nt);
        }
        const float zv = sel ? fmaxf(v, 0.0f) : 0.0f;
        rowPtr[i] = zv;
        if (sel) {
            const uint32_t pos = atomicAdd(&sOut, 1u);
            idxList[(size_t)row * TOPK + pos] = (uint32_t)i;
            valList[(size_t)row * TOPK + pos] = zv;
        }
    }
    __syncthreads();
    if (tid == 0) counts[row] = sOut;
}

// ---------------------------------------------------------------------------
// Sparse decoder: recon[row,:] = sum_i val_i * WdT[idx_i, :]
// One block per row; 256 coalesced 4KB reads of L2-resident WdT.
// ---------------------------------------------------------------------------
__global__ __launch_bounds__(256) void k_decoder(const float* __restrict__ WdT,
                                                 const uint32_t* __restrict__ idxList,
                                                 const float* __restrict__ valList,
                                                 const uint32_t* __restrict__ counts,
                                                 float* __restrict__ recon) {
    __shared__ uint32_t sIdx[TOPK];
    __shared__ float    sVal[TOPK];
    const int tid = threadIdx.x;
    const int row = blockIdx.x;
    const uint32_t cnt = counts[row];
    if (tid < TOPK) {
        sIdx[tid] = idxList[(size_t)row * TOPK + tid];
        sVal[tid] = valList[(size_t)row * TOPK + tid];
    }
    __syncthreads();

    float4 acc = {0.0f, 0.0f, 0.0f, 0.0f};
    const int d0 = tid * 4;
    for (uint32_t i = 0; i < cnt; ++i) {
        const float  s = sVal[i];
        const float4 w = *(const float4*)&WdT[(size_t)sIdx[i] * DM + d0];
        acc.x += s * w.x; acc.y += s * w.y;
        acc.z += s * w.z; acc.w += s * w.w;
    }
    *(float4*)&recon[(size_t)row * DM + d0] = acc;
}

// ---------------------------------------------------------------------------
extern "C" void kernel_launch(void* const* d_in, const int* in_sizes, int n_in,
                              void* d_out, int out_size, void* d_ws, size_t ws_size,
                              hipStream_t stream) {
    const float* x     = (const float*)d_in[0];   // [8192, 1024]
    const float* W_enc = (const float*)d_in[1];   // [16384, 1024]
    const float* b_enc = (const float*)d_in[2];   // [16384]
    const float* W_dec = (const float*)d_in[3];   // [1024, 16384]

    float* recon = (float*)d_out;                       // [8192, 1024]
    float* z     = recon + (size_t)N_ROWS * DM;         // [8192, 16384]

    char* ws = (char*)d_ws;
    float*    WdT     = (float*)ws;                                     // 64MB
    uint32_t* idxList = (uint32_t*)(ws + (size_t)DICT * DM * 4);        // 8MB
    float*    valList = (float*)(ws + (size_t)DICT * DM * 4
                                    + (size_t)N_ROWS * TOPK * 4);       // 8MB
    uint32_t* counts  = (uint32_t*)(ws + (size_t)DICT * DM * 4
                                       + 2 * (size_t)N_ROWS * TOPK * 4);

    dim3 gT(DICT / 32, DM / 32), bT(32, 8);
    k_transpose<<<gT, bT, 0, stream>>>(W_dec, WdT);

    dim3 gE(DICT / 128, N_ROWS / 128);
    k_encoder<<<gE, 256, 0, stream>>>(x, W_enc, b_enc, z);

    k_topk<<<N_ROWS, 512, 0, stream>>>(z, idxList, valList, counts);

    k_decoder<<<N_ROWS, 256, 0, stream>>>(WdT, idxList, valList, counts, recon);
}